// RMLoss_4346506903757
// MI455X (gfx1250) — compile-verified
//
#include <hip/hip_runtime.h>
#include <hip/hip_bf16.h>

typedef __attribute__((ext_vector_type(2))) float v2f;
typedef __attribute__((ext_vector_type(8))) float v8f;

#define BETA 0.001f
#define LOG2E 1.44269504088896340736f
#define LN2   0.69314718055994530942f

// One block (8 wave32) per segment. Computes per-segment mean pair loss.
__global__ __launch_bounds__(256) void rm_seg_kernel(const float* __restrict__ logits,
                                                     const int* __restrict__ cu,
                                                     float* __restrict__ seg_out) {
    __shared__ float xs[128];   // segment logits (zero-padded)
    __shared__ float sq[128];   // 0.5*BETA*x^2 (zero-padded)
    __shared__ float red[8];

    const int seg = blockIdx.x;
    const int a   = cu[seg];
    const int L   = cu[seg + 1] - a;
    const int tid = threadIdx.x;

    if (tid < 128) {
        float v = (tid < L) ? logits[a + tid] : 0.0f;
        xs[tid] = v;
        sq[tid] = 0.5f * BETA * v * v;
    }
    __syncthreads();

    // Scalarize the wave id so tile dispatch is SALU-only: EXEC provably
    // stays all-ones around the WMMAs (ISA requirement).
    const int wave = __builtin_amdgcn_readfirstlane(tid >> 5);
    const int lane = tid & 31;
    const int half = lane >> 4;       // 0: lanes 0-15, 1: lanes 16-31
    const int l16  = lane & 15;
    const float hsel = (half == 0) ? 1.0f : 0.0f;  // zero K=2,3 contributions

    const int nT = (L + 15) >> 4;     // <= 8 tiles of 16
    float acc = 0.0f;
    int cnt = 0;

    for (int tj = 0; tj < nT; ++tj) {
        for (int ti = 0; ti <= tj; ++ti) {
            if ((cnt++ & 7) != wave) continue;   // scalar branch, EXEC untouched
            const int R = ti << 4;
            const int C = tj << 4;

            // A (16x4 f32): col0 = x_row, col1 = 1         -> lanes 0-15 (K=0,1)
            // B (4x16 f32): row0 = 1,    row1 = -x_col     -> lanes 0-15 (K=0,1)
            v2f A, B, A2, B2;
            A[0]  = hsel * xs[R + l16];  A[1]  = hsel;
            B[0]  = hsel;                B[1]  = -hsel * xs[C + l16];
            // Second WMMA: l2 tile  d2[m][n] = sq_row[m] + sq_col[n]
            A2[0] = hsel * sq[R + l16]; A2[1] = hsel;
            B2[0] = hsel;               B2[1] = hsel * sq[C + l16];

            v8f c0 = {};
            v8f d1 = __builtin_amdgcn_wmma_f32_16x16x4_f32(
                false, A, false, B, (short)0, c0, false, false);   // pos - neg
            v8f d2 = __builtin_amdgcn_wmma_f32_16x16x4_f32(
                false, A2, false, B2, (short)0, c0, false, false); // 0.5b(p^2+n^2)

            const int j = C + l16;                 // n = lane & 15
            #pragma unroll
            for (int k = 0; k < 8; ++k) {
                const int i = R + k + (half << 3); // D layout: m = reg + 8*half
                const float d  = d1[k];
                // softplus(-d) = max(-d,0) + ln2 * log2(1 + 2^(-|d|*log2e))
                // raw v_exp_f32 / v_log_f32: arg to log is in (1,2], never denormal
                const float e  = __builtin_amdgcn_exp2f(-fabsf(d) * LOG2E);
                const float sp = __builtin_fmaf(LN2, __builtin_amdgcn_logf(1.0f + e),
                                                fmaxf(-d, 0.0f));
                // Mask as DATA (cndmask of constants), not control flow: keeps the
                // transcendental chain unconditional so it stays flat and batched.
                const float m = ((i < j) && (j < L)) ? 1.0f : 0.0f;
                acc = __builtin_fmaf(m, sp + d2[k], acc);
            }
        }
    }

    // wave32 butterfly reduction
    #pragma unroll
    for (int off = 16; off > 0; off >>= 1)
        acc += __shfl_xor(acc, off, 32);
    if (lane == 0) red[wave] = acc;
    __syncthreads();

    if (tid == 0) {
        float s = 0.0f;
        #pragma unroll
        for (int w = 0; w < 8; ++w) s += red[w];
        const float pairs = 0.5f * (float)L * (float)(L - 1);
        seg_out[seg] = s / pairs;
    }
}

// Deterministic single-block reduction over per-segment means.
__global__ __launch_bounds__(256) void rm_reduce_kernel(const float* __restrict__ seg_out,
                                                        float* __restrict__ out, int nseg) {
    __shared__ float red[256];
    const int tid = threadIdx.x;
    float s = 0.0f;
    for (int i = tid; i < nseg; i += 256) s += seg_out[i];
    red[tid] = s;
    __syncthreads();
    #pragma unroll
    for (int off = 128; off > 0; off >>= 1) {
        if (tid < off) red[tid] += red[tid + off];
        __syncthreads();
    }
    if (tid == 0) out[0] = red[0] / (float)nseg;
}

extern "C" void kernel_launch(void* const* d_in, const int* in_sizes, int n_in,
                              void* d_out, int out_size, void* d_ws, size_t ws_size,
                              hipStream_t stream) {
    const float* logits = (const float*)d_in[0];
    const int*   cu     = (const int*)d_in[1];   // JAX default x64-off -> int32
    const int    nseg   = in_sizes[1] - 1;       // 4096

    float* seg_out = (float*)d_ws;               // nseg floats of scratch

    rm_seg_kernel<<<nseg, 256, 0, stream>>>(logits, cu, seg_out);
    rm_reduce_kernel<<<1, 256, 0, stream>>>(seg_out, (float*)d_out, nseg);
}